// EmbeddingLayer_14516989460967
// MI455X (gfx1250) — compile-verified
//
#include <hip/hip_runtime.h>

// Problem constants (match reference)
#define B_    32
#define C_    2048
#define S_    512
#define DC_   128
#define P_    768
#define OUTD  (DC_ + P_)            // 896 floats per output row
#define ROWS_PER_BLOCK 8            // one wave32 per output row
#define THREADS 256                 // 8 waves

// 1 = stage offsets via gfx1250 GLOBAL_LOAD_ASYNC_TO_LDS_B128 (ASYNCcnt path)
// 0 = plain b128 load + ds_store fallback
#define USE_ASYNC_LDS 1

typedef float v4f __attribute__((ext_vector_type(4)));
typedef int   v4i __attribute__((ext_vector_type(4)));

__global__ __launch_bounds__(THREADS)
void embed_concat_kernel(const int*   __restrict__ char_seq,    // [B*C]
                         const int*   __restrict__ offsets,     // [B*S*2] (start,end) pairs
                         const float* __restrict__ char_table,  // [VOCAB*DC]
                         const float* __restrict__ pretrained,  // [B*S*P]
                         float*       __restrict__ out)         // [B*C*OUTD]
{
    __shared__ int sh_pairs[S_ * 2];                 // raw (start,end) pairs, 4KB

    const int tid        = threadIdx.x;
    const int block_row0 = blockIdx.x * ROWS_PER_BLOCK;
    const int b          = block_row0 / C_;          // uniform: all 8 rows share batch index

    // ---- stage this batch's offset table into LDS: 1024 ints = 4KB,
    //      256 threads x 16B each.
    {
        const int* gsrc = offsets + (size_t)b * (S_ * 2) + 4 * tid;
#if USE_ASYNC_LDS
        // gfx1250 async global->LDS DMA: dst = LDS byte address (VGPR),
        // addr = 64-bit global address (VGPR pair). Tracked on ASYNCcnt.
        unsigned lds_dst = (unsigned)(size_t)(&sh_pairs[4 * tid]);
        unsigned long long gaddr = (unsigned long long)(size_t)gsrc;
        asm volatile("global_load_async_to_lds_b128 %0, %1, off"
                     :: "v"(lds_dst), "v"(gaddr)
                     : "memory");
        asm volatile("s_wait_asynccnt 0x0" ::: "memory");
#else
        v4i o = *(const v4i*)gsrc;
        *(v4i*)(&sh_pairs[4 * tid]) = o;
#endif
    }
    __syncthreads();

    const int wave = tid >> 5;
    const int lane = tid & 31;
    const int row  = block_row0 + wave;              // flat (b,c) row index
    const int c    = row & (C_ - 1);

    // ---- part 1: char embedding gather, 128 floats = 32 x float4 (one per lane)
    const int tok = char_seq[row];
    const v4f* tab4 = (const v4f*)char_table + (size_t)tok * (DC_ / 4);
    v4f* out4 = (v4f*)out + (size_t)row * (OUTD / 4);

    // Speculative near-cache prefetch of the probable pretrained row (contiguous
    // width-4 spans => p == c/4) while the exact LDS search resolves.
    __builtin_prefetch(pretrained + ((size_t)b * S_ + (c >> 2)) * P_, 0, 3);

    v4f ce = tab4[lane];
    __builtin_nontemporal_store(ce, out4 + lane);    // NT: keep L2 for the tables

    // ---- part 2: searchsorted(starts, c, side='right') - 1, clamped (LDS search)
    int lo = 0, hi = S_;
    while (lo < hi) {
        int mid = (lo + hi) >> 1;                    // mid <= 511, always in-bounds
        if (sh_pairs[2 * mid] <= c) lo = mid + 1; else hi = mid;
    }
    int idx = lo - 1;
    idx = idx < 0 ? 0 : (idx > S_ - 1 ? S_ - 1 : idx);
    const bool valid = (c >= sh_pairs[2 * idx]) && (c < sh_pairs[2 * idx + 1]);

    // ---- part 3: broadcast pretrained[b, idx, :] (768 floats = 192 x float4)
    const v4f* src4 = (const v4f*)pretrained + ((size_t)b * S_ + idx) * (P_ / 4);
    v4f* dst4 = out4 + (DC_ / 4);
    const v4f z = {0.f, 0.f, 0.f, 0.f};
#pragma unroll
    for (int i = 0; i < 6; ++i) {                    // 32 lanes x 6 = 192 float4
        v4f v = valid ? src4[lane + 32 * i] : z;
        __builtin_nontemporal_store(v, dst4 + lane + 32 * i);
    }
}

extern "C" void kernel_launch(void* const* d_in, const int* in_sizes, int n_in,
                              void* d_out, int out_size, void* d_ws, size_t ws_size,
                              hipStream_t stream) {
    const int*   char_seq   = (const int*)d_in[0];    // char_sequences [B,C] int32
    const int*   offsets    = (const int*)d_in[1];    // offsets [B,S,2] int32
    const float* char_table = (const float*)d_in[2];  // [VOCAB,DC] fp32
    const float* pretrained = (const float*)d_in[3];  // [B,S,P] fp32
    float*       out        = (float*)d_out;          // [B,C,DC+P] fp32

    dim3 grid((B_ * C_) / ROWS_PER_BLOCK);            // 8192 blocks
    dim3 block(THREADS);                              // 8 wave32s
    hipLaunchKernelGGL(embed_concat_kernel, grid, block, 0, stream,
                       char_seq, offsets, char_table, pretrained, out);
}